// SparseAttention_54099408060778
// MI455X (gfx1250) — compile-verified
//
#include <hip/hip_runtime.h>
#include <math.h>

// Problem constants (from reference): B=4, M=4096, N=4096, D=64, R=128
constexpr int B_ = 4, M_ = 4096, N_ = 4096, D_ = 64, R_ = 128;
constexpr int ROWP = D_ + 4;   // padded LDS row stride (floats): 68*4B=272B, 16B-aligned,
                               // bank shift of 4/row -> conflict-free WMMA A reads

typedef __attribute__((ext_vector_type(2))) float v2f;
typedef __attribute__((ext_vector_type(8))) float v8f;

// CDNA5 async L2->LDS gather (tracked on ASYNCcnt, no VGPR round trip).
// VDST slot = per-lane LDS byte address, VADDR = per-lane 64-bit global address.
__device__ static __forceinline__ void async_gather_b128(unsigned lds_addr,
                                                         unsigned long long gaddr) {
    asm volatile("global_load_async_to_lds_b128 %0, %1, off"
                 :: "v"(lds_addr), "v"(gaddr) : "memory");
}
__device__ static __forceinline__ void wait_async_zero() {
    asm volatile("s_wait_asynccnt 0x0" ::: "memory");
}

// One 128-thread block (4 wave32) per (b, m) query row.
//   Phase 1: async-gather K rows -> LDS, WMMA f32 16x16x4 logits
//            (A = K tile, B = q replicated over columns)
//   Phase 2: issue async V gather (overlaps softmax), softmax over R=128
//   Phase 3: WMMA output (A = V^T tile, B = weights replicated), /sum epilogue
__global__ __launch_bounds__(128) void sparse_attn_wmma_kernel(
    const float* __restrict__ q3d,   // [B, M, D]
    const float* __restrict__ k3d,   // [B, N, D]
    const float* __restrict__ v3d,   // [B, N, D]
    const float* __restrict__ mask,  // [M, R]
    const int*   __restrict__ cidx,  // [M, R]
    float*       __restrict__ out)   // [B, M, D]
{
    __shared__ float buf[R_ * ROWP]; // 34816 B: K staging, then reused for V
    __shared__ float qs[D_];
    __shared__ float s_l[R_];        // logits
    __shared__ float w_l[R_];        // exp(logit - max)

    const int bm   = blockIdx.x;     // 0 .. B*M-1
    const int b    = bm / M_;
    const int m    = bm - b * M_;
    const int t    = threadIdx.x;    // 0..127
    const int wave = t >> 5;         // 0..3
    const int lane = t & 31;
    const int half = lane >> 4;      // 0: lanes 0-15, 1: lanes 16-31
    const int lr   = lane & 15;

    // Per-thread gathered row index (block-uniform m -> thread t owns row t).
    const int ci = cidx[m * R_ + t];

    // LDS byte address of this thread's staging row (flat-LDS aperture: low 32 bits
    // of a generic shared pointer are the DS byte offset).
    const unsigned lrow = (unsigned)(size_t)(const void*)(buf + t * ROWP);

    // ---- Phase 1a: async-gather 128 K rows directly into LDS ----
    {
        const unsigned long long gk =
            (unsigned long long)(size_t)(k3d + ((size_t)b * N_ + (size_t)ci) * D_);
        #pragma unroll
        for (int i = 0; i < D_ / 4; ++i)
            async_gather_b128(lrow + 16u * i, gk + 16ull * i);
    }
    // Overlap K gather latency with q staging.
    if (t < D_) qs[t] = q3d[((size_t)(b * M_ + m)) * (size_t)D_ + t];
    wait_async_zero();
    __syncthreads();

    // ---- Phase 1b: logits via V_WMMA_F32_16X16X4_F32 ----
    // Wave w handles r-tiles 2w and 2w+1. A[i][k]=Kg[r0+i][c+k] (documented 16x4 layout),
    // B[k][j]=q[c+k] replicated over columns -> C[i][j] = s[r0+i] for all j.
    #pragma unroll
    for (int rt = 0; rt < 2; ++rt) {
        const int r0 = (2 * wave + rt) * 16;
        // lane<16: row r0+lr, K = c+0,c+1 ; lane>=16: row r0+lr, K = c+2,c+3
        const float* arow = buf + (r0 + lr) * ROWP + 2 * half;
        v8f acc = {};
        #pragma unroll
        for (int c = 0; c < D_; c += 4) {
            v2f a, bb;
            a.x  = arow[c];
            a.y  = arow[c + 1];
            bb.x = qs[c + 2 * half];
            bb.y = qs[c + 2 * half + 1];
            acc = __builtin_amdgcn_wmma_f32_16x16x4_f32(
                false, a, false, bb, (short)0, acc, false, false);
        }
        // C layout: VGPR g = rows M=g (lanes 0-15) / M=8+g (lanes 16-31); all columns equal.
        if (lr == 0) {
            #pragma unroll
            for (int g = 0; g < 8; ++g) {
                const int r = r0 + 8 * half + g;
                s_l[r] = acc[g] + mask[m * R_ + r];
            }
        }
    }
    __syncthreads();   // all waves done reading K from buf; s_l complete

    // ---- Phase 2a: kick off async V gather into the freed buffer NOW;
    //      it runs concurrently with the entire softmax phase. ----
    {
        const unsigned long long gv =
            (unsigned long long)(size_t)(v3d + ((size_t)b * N_ + (size_t)ci) * D_);
        #pragma unroll
        for (int i = 0; i < D_ / 4; ++i)
            async_gather_b128(lrow + 16u * i, gv + 16ull * i);
    }

    // ---- Phase 2b: softmax over R=128 (sum folded into epilogue) ----
    float mx = -3.402823466e38f;
    for (int r = 0; r < R_; ++r) mx = fmaxf(mx, s_l[r]);
    w_l[t] = expf(s_l[t] - mx);          // t covers 0..127 == R_
    __syncthreads();
    float sum = 0.0f;
    for (int r = 0; r < R_; ++r) sum += w_l[r];
    const float inv_sum = 1.0f / sum;

    wait_async_zero();   // V rows landed in LDS
    __syncthreads();

    // ---- Phase 3: output via WMMA. Wave w owns d-tile d0=16w.
    // A[i][k] = Vg[kc+k][d0+i] (V^T tile), B[k][j] = w[kc+k] replicated
    //  -> C[i][j] = sum_k w[kc+k] * Vg[kc+k][d0+i]; row 0 of C = out[d0+j].
    {
        const int d0 = 16 * wave;
        v8f acc = {};
        #pragma unroll 4
        for (int kc = 0; kc < R_; kc += 4) {
            const int kb = kc + 2 * half;   // rows kb, kb+1 for this lane half
            v2f a, bb;
            a.x  = buf[(kb + 0) * ROWP + d0 + lr];
            a.y  = buf[(kb + 1) * ROWP + d0 + lr];
            bb.x = w_l[kb + 0];
            bb.y = w_l[kb + 1];
            acc = __builtin_amdgcn_wmma_f32_16x16x4_f32(
                false, a, false, bb, (short)0, acc, false, false);
        }
        if (lane < 16) {
            out[((size_t)(b * M_ + m)) * (size_t)D_ + d0 + lane] = acc[0] * inv_sum;
        }
    }
}

extern "C" void kernel_launch(void* const* d_in, const int* in_sizes, int n_in,
                              void* d_out, int out_size, void* d_ws, size_t ws_size,
                              hipStream_t stream) {
    const float* q3d  = (const float*)d_in[0];
    const float* k3d  = (const float*)d_in[1];
    const float* v3d  = (const float*)d_in[2];
    const float* mask = (const float*)d_in[3];
    const int*   cidx = (const int*)d_in[4];
    float* out = (float*)d_out;

    dim3 grid(B_ * M_);   // 16384 blocks, one per (b, m)
    dim3 block(128);      // 4 wave32
    sparse_attn_wmma_kernel<<<grid, block, 0, stream>>>(q3d, k3d, v3d, mask, cidx, out);
}